// BoundaryTransformerLayer_20890720928294
// MI455X (gfx1250) — compile-verified
//
#include <hip/hip_runtime.h>
#include <hip/hip_bf16.h>
#include <math.h>

typedef __attribute__((ext_vector_type(2))) float v2f;
typedef __attribute__((ext_vector_type(8))) float v8f;

#define NS   16
#define CIN  64
#define CMID 64
#define CWS  8
#define BN_EPS 1e-5f

// -------------------------------------------------------------------------
// 0. zero the stats accumulators (150 floats)
// -------------------------------------------------------------------------
__global__ void init_stats_kernel(float* __restrict__ stats, int count) {
    int t = blockIdx.x * blockDim.x + threadIdx.x;
    if (t < count) stats[t] = 0.0f;
}

// -------------------------------------------------------------------------
// 1. q/k/v GEMMs: out = x @ W.T + b   (n x 64) = (n x 64)(64 x 64)
//    WMMA f32 16x16x4. One wave -> 16 rows x 64 cols (4 accumulator tiles).
//    A frag (16x4): lanes 0-15: M=lane, K={kk,kk+1}; lanes 16-31: M=lane-16,
//    K={kk+2,kk+3}.  B frag (4x16): v0 holds rows {kk,kk+2}, v1 {kk+1,kk+3},
//    col = lane&15.  B[k][col] = W[col*64 + k] -> contiguous float2.
// -------------------------------------------------------------------------
__global__ void gemm_qkv_kernel(const float* __restrict__ x,
                                const float* __restrict__ Wq, const float* __restrict__ bq,
                                const float* __restrict__ Wk, const float* __restrict__ bk,
                                const float* __restrict__ Wv, const float* __restrict__ bv,
                                float* __restrict__ xq, float* __restrict__ xk,
                                float* __restrict__ xv, int n) {
    const int lane   = threadIdx.x & 31;
    const int wave   = threadIdx.x >> 5;
    const int wavesB = blockDim.x >> 5;
    const int rowBase = (blockIdx.x * wavesB + wave) * 16;
    if (rowBase >= n) return;  // wave-uniform exit: EXEC stays all-ones for WMMA

    const float* W; const float* bias; float* out;
    if (blockIdx.y == 0)      { W = Wq; bias = bq; out = xq; }
    else if (blockIdx.y == 1) { W = Wk; bias = bk; out = xk; }
    else                      { W = Wv; bias = bv; out = xv; }

    v8f acc0 = {0.f,0.f,0.f,0.f,0.f,0.f,0.f,0.f};
    v8f acc1 = acc0, acc2 = acc0, acc3 = acc0;

    const int m     = lane & 15;
    const int khalf = (lane >> 4) * 2;     // 0 for lanes 0-15, 2 for lanes 16-31
    const float* xrow = x + (size_t)(rowBase + m) * CIN;

    #pragma unroll
    for (int kk = 0; kk < CIN; kk += 4) {
        v2f a = *(const v2f*)(xrow + kk + khalf);
        v2f b0 = *(const v2f*)(W + (size_t)(0 * 16 + m) * CIN + kk + khalf);
        v2f b1 = *(const v2f*)(W + (size_t)(1 * 16 + m) * CIN + kk + khalf);
        v2f b2 = *(const v2f*)(W + (size_t)(2 * 16 + m) * CIN + kk + khalf);
        v2f b3 = *(const v2f*)(W + (size_t)(3 * 16 + m) * CIN + kk + khalf);
        acc0 = __builtin_amdgcn_wmma_f32_16x16x4_f32(false, a, false, b0, (short)0, acc0, false, false);
        acc1 = __builtin_amdgcn_wmma_f32_16x16x4_f32(false, a, false, b1, (short)0, acc1, false, false);
        acc2 = __builtin_amdgcn_wmma_f32_16x16x4_f32(false, a, false, b2, (short)0, acc2, false, false);
        acc3 = __builtin_amdgcn_wmma_f32_16x16x4_f32(false, a, false, b3, (short)0, acc3, false, false);
    }

    // C/D layout: VGPR v -> row = v (lanes 0-15) or v+8 (lanes 16-31); col = lane&15
    const int rAdd = (lane >> 4) * 8;
    #pragma unroll
    for (int v = 0; v < 8; ++v) {
        int row = rowBase + v + rAdd;
        float* orow = out + (size_t)row * CMID;
        orow[0 * 16 + m] = acc0[v] + bias[0 * 16 + m];
        orow[1 * 16 + m] = acc1[v] + bias[1 * 16 + m];
        orow[2 * 16 + m] = acc2[v] + bias[2 * 16 + m];
        orow[3 * 16 + m] = acc3[v] + bias[3 * 16 + m];
    }
}

// -------------------------------------------------------------------------
// 2. stats for bn_p: t = (p[idx]-p) @ Wp1.T + bp1, 3 channels over n*ns rows
// -------------------------------------------------------------------------
__global__ void stats_p_kernel(const float* __restrict__ p, const int* __restrict__ idx,
                               const float* __restrict__ Wp1, const float* __restrict__ bp1,
                               float* __restrict__ stats, int total) {
    float s[3] = {0.f,0.f,0.f}, q[3] = {0.f,0.f,0.f};
    for (int e = blockIdx.x * blockDim.x + threadIdx.x; e < total;
         e += gridDim.x * blockDim.x) {
        int i  = e >> 4;           // ns == 16
        int nb = idx[e];
        float g0 = p[nb * 3 + 0] - p[i * 3 + 0];
        float g1 = p[nb * 3 + 1] - p[i * 3 + 1];
        float g2 = p[nb * 3 + 2] - p[i * 3 + 2];
        #pragma unroll
        for (int c = 0; c < 3; ++c) {
            float t = g0 * Wp1[c * 3 + 0] + g1 * Wp1[c * 3 + 1] + g2 * Wp1[c * 3 + 2] + bp1[c];
            s[c] += t; q[c] += t * t;
        }
    }
    __shared__ float sh[6];
    if (threadIdx.x < 6) sh[threadIdx.x] = 0.f;
    __syncthreads();
    #pragma unroll
    for (int c = 0; c < 3; ++c) { atomicAdd(&sh[c], s[c]); atomicAdd(&sh[3 + c], q[c]); }
    __syncthreads();
    if (threadIdx.x < 6) atomicAdd(&stats[threadIdx.x], sh[threadIdx.x]);
}

// -------------------------------------------------------------------------
// finalize BN: params[c] = gamma*rsqrt(var+eps);  params[C+c] = beta - mean*scale
// -------------------------------------------------------------------------
__global__ void finalize_bn_kernel(const float* __restrict__ stats,
                                   const float* __restrict__ gamma,
                                   const float* __restrict__ beta,
                                   float* __restrict__ params, int C, float invCount) {
    int c = threadIdx.x;
    if (c < C) {
        float mean = stats[c] * invCount;
        float var  = stats[C + c] * invCount - mean * mean;
        float sc   = rsqrtf(var + BN_EPS) * gamma[c];
        params[c]     = sc;
        params[C + c] = beta[c] - mean * sc;
    }
}

// -------------------------------------------------------------------------
// 3. p_r / w_pre / v_plus + stats for bn_w0 (64 ch).
//    64 threads own one (i,j) pair; groups loop over PAIRS_PER_GROUP pairs.
// -------------------------------------------------------------------------
#define PAIRS_PER_GROUP 16
__global__ void pr_wpre_kernel(const float* __restrict__ p, const int* __restrict__ idx,
                               const float* __restrict__ Wp1, const float* __restrict__ bp1,
                               const float* __restrict__ bnp,   // 6: scale[3], shift[3]
                               const float* __restrict__ Wp2, const float* __restrict__ bp2,
                               const float* __restrict__ xq, const float* __restrict__ xk,
                               const float* __restrict__ xv,
                               float* __restrict__ w_pre, float* __restrict__ v_plus,
                               float* __restrict__ stats_w0, int total) {
    const int cc     = threadIdx.x & 63;
    const int group  = threadIdx.x >> 6;
    const int groups = blockDim.x >> 6;
    const int base   = (blockIdx.x * groups + group) * PAIRS_PER_GROUP;

    float w2c0 = Wp2[cc * 3 + 0], w2c1 = Wp2[cc * 3 + 1], w2c2 = Wp2[cc * 3 + 2];
    float bcc = bp2[cc];
    float s = 0.f, q = 0.f;

    for (int t = 0; t < PAIRS_PER_GROUP; ++t) {
        int e = base + t;
        if (e >= total) break;        // group-uniform
        int i  = e >> 4;              // ns == 16
        int nb = idx[e];
        float g0 = p[nb * 3 + 0] - p[i * 3 + 0];
        float g1 = p[nb * 3 + 1] - p[i * 3 + 1];
        float g2 = p[nb * 3 + 2] - p[i * 3 + 2];
        float pr3[3];
        #pragma unroll
        for (int c = 0; c < 3; ++c) {
            float tt = g0 * Wp1[c * 3 + 0] + g1 * Wp1[c * 3 + 1] + g2 * Wp1[c * 3 + 2] + bp1[c];
            pr3[c] = fmaxf(tt * bnp[c] + bnp[3 + c], 0.f);
        }
        float pr = pr3[0] * w2c0 + pr3[1] * w2c1 + pr3[2] * w2c2 + bcc;
        float wv = xk[(size_t)nb * CMID + cc] - xq[(size_t)i * CMID + cc] + pr;
        w_pre [(size_t)e * CMID + cc] = wv;
        v_plus[(size_t)e * CMID + cc] = xv[(size_t)nb * CMID + cc] + pr;
        s += wv; q += wv * wv;
    }
    __shared__ float sh[128];
    if (threadIdx.x < 128) sh[threadIdx.x] = 0.f;
    __syncthreads();
    atomicAdd(&sh[cc], s);
    atomicAdd(&sh[64 + cc], q);
    __syncthreads();
    if (threadIdx.x < 128) atomicAdd(&stats_w0[threadIdx.x], sh[threadIdx.x]);
}

// -------------------------------------------------------------------------
// 4. w1 = relu(bn64(w_pre)) @ Ww1.T + bw1 (8 outputs) + stats for bn_w1
// -------------------------------------------------------------------------
__global__ void w1_kernel(const float* __restrict__ w_pre,
                          const float* __restrict__ bn0,   // 128: scale[64], shift[64]
                          const float* __restrict__ Ww1, const float* __restrict__ bw1,
                          float* __restrict__ w1, float* __restrict__ stats_w1, int total) {
    __shared__ float shW[CWS * CMID];   // 512
    __shared__ float shB[2 * CMID];     // 128
    __shared__ float shb1[CWS];
    for (int t = threadIdx.x; t < CWS * CMID; t += blockDim.x) shW[t] = Ww1[t];
    for (int t = threadIdx.x; t < 2 * CMID; t += blockDim.x)   shB[t] = bn0[t];
    if (threadIdx.x < CWS) shb1[threadIdx.x] = bw1[threadIdx.x];
    __syncthreads();

    float s[CWS], q[CWS];
    #pragma unroll
    for (int o = 0; o < CWS; ++o) { s[o] = 0.f; q[o] = 0.f; }

    for (int e = blockIdx.x * blockDim.x + threadIdx.x; e < total;
         e += gridDim.x * blockDim.x) {
        const float* wp = w_pre + (size_t)e * CMID;
        float r[CWS];
        #pragma unroll
        for (int o = 0; o < CWS; ++o) r[o] = shb1[o];
        #pragma unroll
        for (int k = 0; k < CMID; ++k) {
            float v = fmaxf(wp[k] * shB[k] + shB[CMID + k], 0.f);
            #pragma unroll
            for (int o = 0; o < CWS; ++o) r[o] += v * shW[o * CMID + k];
        }
        #pragma unroll
        for (int o = 0; o < CWS; ++o) {
            w1[(size_t)e * CWS + o] = r[o];
            s[o] += r[o]; q[o] += r[o] * r[o];
        }
    }
    __shared__ float sh[2 * CWS];
    if (threadIdx.x < 2 * CWS) sh[threadIdx.x] = 0.f;
    __syncthreads();
    #pragma unroll
    for (int o = 0; o < CWS; ++o) { atomicAdd(&sh[o], s[o]); atomicAdd(&sh[CWS + o], q[o]); }
    __syncthreads();
    if (threadIdx.x < 2 * CWS) atomicAdd(&stats_w1[threadIdx.x], sh[threadIdx.x]);
}

// -------------------------------------------------------------------------
// 5. w2 = relu(bn8(w1)) @ Ww2.T + bw2 ; softmax over neighbors; weighted sum.
//    One thread per (point, channel c in 0..7).
// -------------------------------------------------------------------------
__global__ void out_kernel(const float* __restrict__ w1,
                           const float* __restrict__ bn1,  // 16: scale[8], shift[8]
                           const float* __restrict__ Ww2, const float* __restrict__ bw2,
                           const float* __restrict__ v_plus,
                           float* __restrict__ out, int n) {
    int t = blockIdx.x * blockDim.x + threadIdx.x;
    int i = t >> 3, c = t & 7;
    if (i >= n) return;

    float wv[NS];
    float mx = -INFINITY;
    #pragma unroll
    for (int j = 0; j < NS; ++j) {
        const float* w1p = w1 + ((size_t)i * NS + j) * CWS;
        float r = bw2[c];
        #pragma unroll
        for (int k = 0; k < CWS; ++k)
            r += fmaxf(w1p[k] * bn1[k] + bn1[CWS + k], 0.f) * Ww2[c * CWS + k];
        wv[j] = r;
        mx = fmaxf(mx, r);
    }
    float denom = 0.f;
    #pragma unroll
    for (int j = 0; j < NS; ++j) { wv[j] = expf(wv[j] - mx); denom += wv[j]; }
    float inv = 1.f / denom;

    float o[CWS];
    #pragma unroll
    for (int sj = 0; sj < CWS; ++sj) o[sj] = 0.f;
    #pragma unroll
    for (int j = 0; j < NS; ++j) {
        float wj = wv[j] * inv;
        const float* vp = v_plus + ((size_t)i * NS + j) * CMID;
        #pragma unroll
        for (int sj = 0; sj < CWS; ++sj) o[sj] += vp[sj * CWS + c] * wj;
    }
    #pragma unroll
    for (int sj = 0; sj < CWS; ++sj) out[(size_t)i * CMID + sj * CWS + c] = o[sj];
}

// -------------------------------------------------------------------------
// launch
// -------------------------------------------------------------------------
extern "C" void kernel_launch(void* const* d_in, const int* in_sizes, int n_in,
                              void* d_out, int out_size, void* d_ws, size_t ws_size,
                              hipStream_t stream) {
    const float* p     = (const float*)d_in[0];
    const float* x     = (const float*)d_in[1];
    const int*   idx   = (const int*)  d_in[2];
    const float* Wq    = (const float*)d_in[3];
    const float* bq    = (const float*)d_in[4];
    const float* Wk    = (const float*)d_in[5];
    const float* bk    = (const float*)d_in[6];
    const float* Wv    = (const float*)d_in[7];
    const float* bv    = (const float*)d_in[8];
    const float* Wp1   = (const float*)d_in[9];
    const float* bp1   = (const float*)d_in[10];
    const float* bnp_g = (const float*)d_in[11];
    const float* bnp_b = (const float*)d_in[12];
    const float* Wp2   = (const float*)d_in[13];
    const float* bp2   = (const float*)d_in[14];
    const float* bn0_g = (const float*)d_in[15];
    const float* bn0_b = (const float*)d_in[16];
    const float* Ww1   = (const float*)d_in[17];
    const float* bw1   = (const float*)d_in[18];
    const float* bn1_g = (const float*)d_in[19];
    const float* bn1_b = (const float*)d_in[20];
    const float* Ww2   = (const float*)d_in[21];
    const float* bw2   = (const float*)d_in[22];
    float* out = (float*)d_out;

    const int n     = in_sizes[0] / 3;     // 65536
    const int total = n * NS;              // n * 16
    const float invCount = 1.0f / (float)total;

    // ---- workspace layout (floats) ----
    float* ws = (float*)d_ws;
    size_t off = 0;
    float* xq = ws + off; off += (size_t)n * CMID;
    float* xk = ws + off; off += (size_t)n * CMID;
    float* xv = ws + off; off += (size_t)n * CMID;
    float* stats = ws + off; off += 160;   // [0..5]=p, [6..133]=w0, [134..149]=w1
    float* stats_p  = stats;
    float* stats_w0 = stats + 6;
    float* stats_w1 = stats + 134;
    float* params = ws + off; off += 160;  // [0..5]=p, [6..133]=w0, [134..149]=w1
    float* bnp = params;
    float* bn0 = params + 6;
    float* bn1 = params + 134;
    float* w_pre  = ws + off; off += (size_t)total * CMID;
    float* v_plus = ws + off; off += (size_t)total * CMID;
    float* w1     = ws + off; off += (size_t)total * CWS;
    (void)ws_size; (void)n_in; (void)out_size;

    // 0. zero stats
    init_stats_kernel<<<1, 256, 0, stream>>>(stats, 150);

    // 1. q/k/v GEMMs via WMMA: 8 waves/block -> 128 rows/block
    {
        dim3 grid((n + 127) / 128, 3, 1);
        gemm_qkv_kernel<<<grid, 256, 0, stream>>>(x, Wq, bq, Wk, bk, Wv, bv,
                                                  xq, xk, xv, n);
    }

    // 2. bn_p stats + finalize
    stats_p_kernel<<<1024, 256, 0, stream>>>(p, idx, Wp1, bp1, stats_p, total);
    finalize_bn_kernel<<<1, 64, 0, stream>>>(stats_p, bnp_g, bnp_b, bnp, 3, invCount);

    // 3. p_r / w_pre / v_plus + bn_w0 stats, then finalize
    {
        int pairsPerBlock = (256 / 64) * PAIRS_PER_GROUP;  // 64
        int grid = (total + pairsPerBlock - 1) / pairsPerBlock;
        pr_wpre_kernel<<<grid, 256, 0, stream>>>(p, idx, Wp1, bp1, bnp, Wp2, bp2,
                                                 xq, xk, xv, w_pre, v_plus,
                                                 stats_w0, total);
    }
    finalize_bn_kernel<<<1, 64, 0, stream>>>(stats_w0, bn0_g, bn0_b, bn0, CMID, invCount);

    // 4. w1 + bn_w1 stats, then finalize
    w1_kernel<<<2048, 256, 0, stream>>>(w_pre, bn0, Ww1, bw1, w1, stats_w1, total);
    finalize_bn_kernel<<<1, 64, 0, stream>>>(stats_w1, bn1_g, bn1_b, bn1, CWS, invCount);

    // 5. w2 + softmax + weighted aggregation
    {
        int threads = n * CWS;
        out_kernel<<<(threads + 255) / 256, 256, 0, stream>>>(w1, bn1, Ww2, bw2,
                                                              v_plus, out, n);
    }
}